// ParaphraseVAE_26104811225138
// MI455X (gfx1250) — compile-verified
//
#include <hip/hip_runtime.h>
#include <hip/hip_bf16.h>
#include <math.h>

// ---------------------------------------------------------------------------
// ParaphraseVAE forward for MI455X (gfx1250, wave32).
//   K1: encoder bidirectional GRU scan   (1 block x 1024, sequential, 48 steps)
//   K2: latent mean/logv/z/latent_out    (1 block x 1024)
//   K3: decoder attention+GRU scan       (1 block x 1024, sequential, 50 steps)
//       -> stages hidden states into Hdec, K-pair-major (k/2, n*2+k&1), 600x64
//   K4: vocab projection out_w(50000x600) x Hdec(600x64):
//       - Hdec staged to LDS via GLOBAL_LOAD_ASYNC_TO_LDS_B128 (ASYNCcnt)
//       - V_WMMA_F32_16X16X4_F32, B fragments via single ds_load_b64
//       - out_w streamed once (120 MB @ 23.3 TB/s ~ 5.2 us roofline)
//   K5: log_softmax over V per timestep (in place on d_out)
// ---------------------------------------------------------------------------

#define Vv   50000
#define Hh   600
#define HHh  300
#define Ll   100
#define MAXL 50
#define TIN  48
#define TTGT 50
#define NPAD 64           // padded time dim for WMMA N tiles
#define NT   1024         // threads for recurrent kernels
#define HDEC_FLOATS (Hh * NPAD)          // 38400
#define HDEC_BYTES  (HDEC_FLOATS * 4)    // 153600

typedef __attribute__((ext_vector_type(2))) float v2f;
typedef __attribute__((ext_vector_type(8))) float v8f;

__device__ __forceinline__ float sigf(float x) { return 1.0f / (1.0f + expf(-x)); }

__device__ __forceinline__ float wave_sum(float v) {
  #pragma unroll
  for (int off = 16; off > 0; off >>= 1) v += __shfl_xor(v, off, 32);
  return v;
}

// Dual-accumulator packed-f32 dot product; n must be a multiple of 4 and both
// pointers 8B-aligned. Lowers to v_pk_fma_f32 chains after inlining.
__device__ __forceinline__ float dot_v2(const float* __restrict__ w,
                                        const float* __restrict__ x, int n) {
  const v2f* wv = (const v2f*)w;
  const v2f* xv = (const v2f*)x;
  v2f a0 = {}, a1 = {};
  const int n2 = n >> 1;
  for (int i = 0; i < n2; i += 2) {
    a0 += wv[i] * xv[i];
    a1 += wv[i + 1] * xv[i + 1];
  }
  return a0.x + a0.y + a1.x + a1.y;
}

// ---------------------------------------------------------------- encoder ---
__global__ __launch_bounds__(NT) void encoder_kernel(
    const float* __restrict__ emb_enc, const int* __restrict__ toks,
    const float* __restrict__ wih_f, const float* __restrict__ whh_f,
    const float* __restrict__ bih_f, const float* __restrict__ bhh_f,
    const float* __restrict__ wih_b, const float* __restrict__ whh_b,
    const float* __restrict__ bih_b, const float* __restrict__ bhh_b,
    float* __restrict__ enc_out /* MAXL x H */) {
  const int tid = threadIdx.x;
  __shared__ float s_x[Hh], s_h[Hh];
  __shared__ float s_gi_f[3 * HHh], s_gh_f[3 * HHh];
  __shared__ float s_gi_b[3 * HHh], s_gh_b[3 * HHh];

  for (int i = tid; i < Hh; i += NT) s_h[i] = 0.0f;
  for (int i = tid; i < (MAXL - TIN) * Hh; i += NT) enc_out[TIN * Hh + i] = 0.0f;
  __syncthreads();

  for (int t = 0; t < TIN; ++t) {
    const int tok = toks[t];
    for (int i = tid; i < Hh; i += NT) s_x[i] = emb_enc[(size_t)tok * Hh + i];
    __syncthreads();

    for (int r = tid; r < 3 * HHh; r += NT) {
      s_gi_f[r] = bih_f[r] + dot_v2(wih_f + (size_t)r * Hh, s_x, Hh);
      s_gi_b[r] = bih_b[r] + dot_v2(wih_b + (size_t)r * Hh, s_x, Hh);
      s_gh_f[r] = bhh_f[r] + dot_v2(whh_f + (size_t)r * HHh, s_h, HHh);
      s_gh_b[r] = bhh_b[r] + dot_v2(whh_b + (size_t)r * HHh, s_h + HHh, HHh);
    }
    __syncthreads();

    for (int j = tid; j < HHh; j += NT) {
      float rf = sigf(s_gi_f[j] + s_gh_f[j]);
      float zf = sigf(s_gi_f[HHh + j] + s_gh_f[HHh + j]);
      float nf = tanhf(s_gi_f[2 * HHh + j] + rf * s_gh_f[2 * HHh + j]);
      float hf2 = (1.0f - zf) * nf + zf * s_h[j];
      float rb = sigf(s_gi_b[j] + s_gh_b[j]);
      float zb = sigf(s_gi_b[HHh + j] + s_gh_b[HHh + j]);
      float nb = tanhf(s_gi_b[2 * HHh + j] + rb * s_gh_b[2 * HHh + j]);
      float hb2 = (1.0f - zb) * nb + zb * s_h[HHh + j];
      s_h[j] = hf2; s_h[HHh + j] = hb2;
      enc_out[t * Hh + j] = hf2; enc_out[t * Hh + HHh + j] = hb2;
    }
    __syncthreads();
  }
}

// ------------------------------------------------------------------ latent ---
__global__ __launch_bounds__(NT) void latent_kernel(
    const float* __restrict__ enc_out,
    const float* __restrict__ w_mean, const float* __restrict__ b_mean,
    const float* __restrict__ w_logv, const float* __restrict__ b_logv,
    const float* __restrict__ w_l2h, const float* __restrict__ b_l2h,
    const float* __restrict__ eps,
    float* __restrict__ mean_out, float* __restrict__ logv_out,
    float* __restrict__ latent_out /* MAXL x H */) {
  const int tid = threadIdx.x;
  __shared__ float s_z[MAXL * Ll];

  for (int idx = tid; idx < MAXL * Ll; idx += NT) {
    const int t = idx / Ll, l = idx % Ll;
    const float* e = enc_out + t * Hh;
    const float m = b_mean[l] + dot_v2(w_mean + (size_t)l * Hh, e, Hh);
    const float v = b_logv[l] + dot_v2(w_logv + (size_t)l * Hh, e, Hh);
    mean_out[idx] = m;
    logv_out[idx] = v;
    s_z[idx] = eps[idx] * expf(0.5f * v) + m;
  }
  __syncthreads();

  for (int idx = tid; idx < MAXL * Hh; idx += NT) {
    const int t = idx / Hh, i = idx % Hh;
    latent_out[idx] = b_l2h[i] + dot_v2(w_l2h + (size_t)i * Ll, s_z + t * Ll, Ll);
  }
}

// ----------------------------------------------------------------- decoder ---
__global__ __launch_bounds__(NT) void decoder_kernel(
    const float* __restrict__ emb_dec, const int* __restrict__ tgt,
    const float* __restrict__ wih_f, const float* __restrict__ whh_f,
    const float* __restrict__ bih_f, const float* __restrict__ bhh_f,
    const float* __restrict__ wih_b, const float* __restrict__ whh_b,
    const float* __restrict__ bih_b, const float* __restrict__ bhh_b,
    const float* __restrict__ attn_w, const float* __restrict__ attn_b,
    const float* __restrict__ comb_w, const float* __restrict__ comb_b,
    const float* __restrict__ enc_out, const float* __restrict__ latent_out,
    float* __restrict__ Hdec /* K-pair-major: (k>>1)*128 + t*2 + (k&1) */) {
  const int tid = threadIdx.x;
  const int lane = tid & 31, wave = tid >> 5;  // 32 waves
  __shared__ float s_emb[Hh], s_h[Hh], s_app[Hh], s_o[Hh], s_sc[NPAD];
  __shared__ float s_gi_f[3 * HHh], s_gh_f[3 * HHh];
  __shared__ float s_gi_b[3 * HHh], s_gh_b[3 * HHh];

  for (int i = tid; i < HDEC_FLOATS; i += NT) Hdec[i] = 0.0f;
  for (int i = tid; i < Hh; i += NT) s_h[i] = enc_out[(TIN - 1) * Hh + i];
  __syncthreads();

  for (int t = 0; t < TTGT; ++t) {
    const int tok = (t == 0) ? 1 /*EOS*/ : tgt[t - 1];
    for (int i = tid; i < Hh; i += NT) s_emb[i] = emb_dec[(size_t)tok * Hh + i];
    __syncthreads();

    // attention scores: one row per wave, lane-parallel dot + shfl reduce
    for (int a = wave; a < MAXL; a += 32) {
      const float* wr = attn_w + (size_t)a * (2 * Hh);
      float p = 0.0f;
      for (int i = lane; i < 2 * Hh; i += 32) {
        const float xv = (i < Hh) ? s_emb[i] : s_h[i - Hh];
        p += wr[i] * xv;
      }
      p = wave_sum(p);
      if (lane == 0) s_sc[a] = p + attn_b[a];
    }
    __syncthreads();
    if (tid == 0) {  // tiny softmax over 50 entries
      float mx = s_sc[0];
      for (int a = 1; a < MAXL; ++a) mx = fmaxf(mx, s_sc[a]);
      float sm = 0.0f;
      for (int a = 0; a < MAXL; ++a) { const float e = expf(s_sc[a] - mx); s_sc[a] = e; sm += e; }
      const float inv = 1.0f / sm;
      for (int a = 0; a < MAXL; ++a) s_sc[a] *= inv;
    }
    __syncthreads();

    for (int i = tid; i < Hh; i += NT) {
      float acc = 0.0f;
      for (int a = 0; a < MAXL; ++a) acc += s_sc[a] * latent_out[a * Hh + i];
      s_app[i] = acc;
    }
    __syncthreads();

    // combine: o = relu(comb_w @ [emb; applied] + comb_b)
    for (int r = wave; r < Hh; r += 32) {
      const float* wr = comb_w + (size_t)r * (2 * Hh);
      float p = 0.0f;
      for (int i = lane; i < 2 * Hh; i += 32) {
        const float xv = (i < Hh) ? s_emb[i] : s_app[i - Hh];
        p += wr[i] * xv;
      }
      p = wave_sum(p);
      if (lane == 0) s_o[r] = fmaxf(p + comb_b[r], 0.0f);
    }
    __syncthreads();

    // GRU gates (forward + backward cells), packed-f32 dots
    for (int r = tid; r < 3 * HHh; r += NT) {
      s_gi_f[r] = bih_f[r] + dot_v2(wih_f + (size_t)r * Hh, s_o, Hh);
      s_gi_b[r] = bih_b[r] + dot_v2(wih_b + (size_t)r * Hh, s_o, Hh);
      s_gh_f[r] = bhh_f[r] + dot_v2(whh_f + (size_t)r * HHh, s_h, HHh);
      s_gh_b[r] = bhh_b[r] + dot_v2(whh_b + (size_t)r * HHh, s_h + HHh, HHh);
    }
    __syncthreads();

    for (int j = tid; j < HHh; j += NT) {
      float rf = sigf(s_gi_f[j] + s_gh_f[j]);
      float zf = sigf(s_gi_f[HHh + j] + s_gh_f[HHh + j]);
      float nf = tanhf(s_gi_f[2 * HHh + j] + rf * s_gh_f[2 * HHh + j]);
      float hf2 = (1.0f - zf) * nf + zf * s_h[j];
      float rb = sigf(s_gi_b[j] + s_gh_b[j]);
      float zb = sigf(s_gi_b[HHh + j] + s_gh_b[HHh + j]);
      float nb = tanhf(s_gi_b[2 * HHh + j] + rb * s_gh_b[2 * HHh + j]);
      float hb2 = (1.0f - zb) * nb + zb * s_h[HHh + j];
      s_h[j] = hf2; s_h[HHh + j] = hb2;
      // K-pair-major staging for WMMA B operand
      const int k0 = j, k1 = HHh + j;
      Hdec[(k0 >> 1) * (2 * NPAD) + t * 2 + (k0 & 1)] = hf2;
      Hdec[(k1 >> 1) * (2 * NPAD) + t * 2 + (k1 & 1)] = hb2;
    }
    __syncthreads();
  }
}

// ---------------------------------------------- vocab projection via WMMA ---
// logits[t][v] = sum_k out_w[v][k] * Hdec[k][t] + out_b[v]
// B (Hdec) staged to LDS once per block via async-to-LDS, then ds_load_b64.
__global__ __launch_bounds__(256) void vocab_gemm_kernel(
    const float* __restrict__ out_w, const float* __restrict__ out_b,
    const float* __restrict__ Hdec, float* __restrict__ logits) {
  extern __shared__ float s_b[];  // HDEC_BYTES dynamic LDS (153.6 KB of 320 KB)

  // --- async stage Hdec -> LDS (ASYNCcnt path), all waves participate ------
  {
    const unsigned lds_base = (unsigned)(size_t)(void*)s_b;  // low 32b = LDS offset
    for (unsigned ofs = threadIdx.x * 16u; ofs < (unsigned)HDEC_BYTES;
         ofs += 256u * 16u) {
      const unsigned lo = lds_base + ofs;
      asm volatile("global_load_async_to_lds_b128 %0, %1, %2"
                   :
                   : "v"(lo), "v"(ofs), "s"(Hdec)
                   : "memory");
    }
    asm volatile("s_wait_asynccnt 0x0" ::: "memory");
  }
  __syncthreads();

  const int lane = threadIdx.x & 31;
  const int wave = threadIdx.x >> 5;
  const int tile_m = blockIdx.x * 8 + wave;  // wave-uniform
  if (tile_m < Vv / 16) {                    // uniform guard: EXEC all-1s inside
    const int m0 = tile_m * 16;
    const int nl = lane & 15;
    const int khalf = lane >> 4;             // selects K pair 0/1 within k-step
    const float* arow = out_w + (size_t)(m0 + nl) * Hh + 2 * khalf;

    v8f acc0 = {}, acc1 = {}, acc2 = {}, acc3 = {};
    for (int k = 0; k < Hh; k += 4) {
      const float* ap = arow + k;
      __builtin_prefetch(ap + 64, 0, 0);     // global_prefetch_b8
      const v2f av = *(const v2f*)ap;        // global_load_b64 (A: 16x4 f32)
      const float* bp = s_b + ((k >> 1) + khalf) * (2 * NPAD) + nl * 2;
      const v2f b0 = *(const v2f*)(bp);      // ds_load_b64 each
      const v2f b1 = *(const v2f*)(bp + 32);
      const v2f b2 = *(const v2f*)(bp + 64);
      const v2f b3 = *(const v2f*)(bp + 96);
      acc0 = __builtin_amdgcn_wmma_f32_16x16x4_f32(false, av, false, b0, (short)0, acc0, false, false);
      acc1 = __builtin_amdgcn_wmma_f32_16x16x4_f32(false, av, false, b1, (short)0, acc1, false, false);
      acc2 = __builtin_amdgcn_wmma_f32_16x16x4_f32(false, av, false, b2, (short)0, acc2, false, false);
      acc3 = __builtin_amdgcn_wmma_f32_16x16x4_f32(false, av, false, b3, (short)0, acc3, false, false);
    }

    // D layout: lanes 0-15 -> N=lane, M=e; lanes 16-31 -> N=lane-16, M=e+8
    const int mbase = khalf << 3;
    #pragma unroll
    for (int e = 0; e < 8; ++e) {
      const int v = m0 + mbase + e;
      const float bias = out_b[v];
      logits[(size_t)(nl) * Vv + v] = acc0[e] + bias;
      logits[(size_t)(16 + nl) * Vv + v] = acc1[e] + bias;
      logits[(size_t)(32 + nl) * Vv + v] = acc2[e] + bias;
      if (48 + nl < TTGT) logits[(size_t)(48 + nl) * Vv + v] = acc3[e] + bias;
    }
  }
}

// ------------------------------------------------------------- log_softmax ---
__global__ __launch_bounds__(256) void log_softmax_kernel(float* __restrict__ logits) {
  const int t = blockIdx.x;
  float* row = logits + (size_t)t * Vv;
  __shared__ float red[256];
  const int tid = threadIdx.x;

  float m = -INFINITY;
  for (int i = tid; i < Vv; i += 256) m = fmaxf(m, row[i]);
  red[tid] = m; __syncthreads();
  for (int s = 128; s > 0; s >>= 1) {
    if (tid < s) red[tid] = fmaxf(red[tid], red[tid + s]);
    __syncthreads();
  }
  m = red[0]; __syncthreads();

  float sum = 0.0f;
  for (int i = tid; i < Vv; i += 256) sum += expf(row[i] - m);
  red[tid] = sum; __syncthreads();
  for (int s = 128; s > 0; s >>= 1) {
    if (tid < s) red[tid] += red[tid + s];
    __syncthreads();
  }
  const float lse = m + logf(red[0]);
  __syncthreads();

  for (int i = tid; i < Vv; i += 256) row[i] = row[i] - lse;
}

// ------------------------------------------------------------------ launch ---
extern "C" void kernel_launch(void* const* d_in, const int* in_sizes, int n_in,
                              void* d_out, int out_size, void* d_ws, size_t ws_size,
                              hipStream_t stream) {
  const float* emb_enc  = (const float*)d_in[0];
  const float* emb_dec  = (const float*)d_in[1];
  const float* e_wih_f  = (const float*)d_in[2];
  const float* e_whh_f  = (const float*)d_in[3];
  const float* e_bih_f  = (const float*)d_in[4];
  const float* e_bhh_f  = (const float*)d_in[5];
  const float* e_wih_b  = (const float*)d_in[6];
  const float* e_whh_b  = (const float*)d_in[7];
  const float* e_bih_b  = (const float*)d_in[8];
  const float* e_bhh_b  = (const float*)d_in[9];
  const float* d_wih_f  = (const float*)d_in[10];
  const float* d_whh_f  = (const float*)d_in[11];
  const float* d_bih_f  = (const float*)d_in[12];
  const float* d_bhh_f  = (const float*)d_in[13];
  const float* d_wih_b  = (const float*)d_in[14];
  const float* d_whh_b  = (const float*)d_in[15];
  const float* d_bih_b  = (const float*)d_in[16];
  const float* d_bhh_b  = (const float*)d_in[17];
  const float* w_mean   = (const float*)d_in[18];
  const float* b_mean   = (const float*)d_in[19];
  const float* w_logv   = (const float*)d_in[20];
  const float* b_logv   = (const float*)d_in[21];
  const float* w_l2h    = (const float*)d_in[22];
  const float* b_l2h    = (const float*)d_in[23];
  const float* attn_w   = (const float*)d_in[24];
  const float* attn_b   = (const float*)d_in[25];
  const float* comb_w   = (const float*)d_in[26];
  const float* comb_b   = (const float*)d_in[27];
  const float* out_w    = (const float*)d_in[28];
  const float* out_b    = (const float*)d_in[29];
  const float* eps      = (const float*)d_in[30];
  const int*   in_toks  = (const int*)d_in[31];
  const int*   tgt_toks = (const int*)d_in[32];

  float* ws         = (float*)d_ws;
  float* enc_out    = ws;                         // MAXL*H       = 30000
  float* latent_out = ws + MAXL * Hh;             // MAXL*H       = 30000
  float* Hdec       = ws + 2 * MAXL * Hh;         // HDEC_FLOATS  = 38400

  float* decoded  = (float*)d_out;                // TTGT * V
  float* mean_out = decoded + (size_t)TTGT * Vv;  // MAXL * L
  float* logv_out = mean_out + MAXL * Ll;         // MAXL * L

  encoder_kernel<<<1, NT, 0, stream>>>(emb_enc, in_toks,
      e_wih_f, e_whh_f, e_bih_f, e_bhh_f,
      e_wih_b, e_whh_b, e_bih_b, e_bhh_b, enc_out);

  latent_kernel<<<1, NT, 0, stream>>>(enc_out, w_mean, b_mean, w_logv, b_logv,
      w_l2h, b_l2h, eps, mean_out, logv_out, latent_out);

  decoder_kernel<<<1, NT, 0, stream>>>(emb_dec, tgt_toks,
      d_wih_f, d_whh_f, d_bih_f, d_bhh_f,
      d_wih_b, d_whh_b, d_bih_b, d_bhh_b,
      attn_w, attn_b, comb_w, comb_b, enc_out, latent_out, Hdec);

  const int mtiles = Vv / 16;  // 3125
  vocab_gemm_kernel<<<(mtiles + 7) / 8, 256, HDEC_BYTES, stream>>>(
      out_w, out_b, Hdec, decoded);

  log_softmax_kernel<<<TTGT, 256, 0, stream>>>(decoded);
}